// QLinearGate_65481071408765
// MI455X (gfx1250) — compile-verified
//
#include <hip/hip_runtime.h>
#include <hip/hip_bf16.h>

// out[b,q] = prod_{i<=q} cos(x[b,i]).  Closed form of the 4-qubit circuit:
// RZ phases + CNOT permutations cannot change computational-basis probabilities,
// so `weight` provably does not affect the output, and the CNOT chain turns
// <Z_q> into a product of per-qubit cos(x_i) expectations.
//
// Memory-bound streaming kernel: 8 MB in + 8 MB out -> ~0.72 us at 23.3 TB/s.

typedef float v4f __attribute__((ext_vector_type(4)));
typedef int   v4i __attribute__((ext_vector_type(4)));

#define THREADS 256
#define ITEMS   4                      // float4 rows per thread
#define TILE_BYTES 4096                // THREADS*16: tile stride in BOTH global and LDS

__device__ __forceinline__ float fast_cos(float x) {
    // v_cos_f32 takes revolutions: cos(2*pi*f). x ~ N(0,1) -> |f| < ~1, in range.
    return __builtin_amdgcn_cosf(x * 0.15915494309189535f);
}

__device__ __forceinline__ v4f cumcos(v4f v) {
    float c0 = fast_cos(v.x);
    float c1 = c0 * fast_cos(v.y);
    float c2 = c1 * fast_cos(v.z);
    float c3 = c2 * fast_cos(v.w);
    v4f o = {c0, c1, c2, c3};
    return o;
}

__global__ __launch_bounds__(THREADS) void qexpz_kernel(const v4f* __restrict__ x,
                                                        v4f* __restrict__ out,
                                                        int B) {
    const int tid  = threadIdx.x;
    const int base = blockIdx.x * (THREADS * ITEMS) + tid;

#if defined(__gfx1250__) && __has_builtin(__builtin_amdgcn_global_load_async_to_lds_b128)
    __shared__ v4f smem[ITEMS * THREADS];

    if (base + (ITEMS - 1) * THREADS < B) {
        // Fast path: whole tile in range (always true for B % 1024 == 0).
        // INST_OFFSET is added to both the global and the LDS address, and our
        // LDS tile stride equals the global tile stride (4096 B), so all four
        // async loads share one VADDR/VDST pair and differ only in imm offset.
        // Offsets must be literal constants (Ii immediates) -> manual unroll.
        v4i* gsrc = (v4i*)(x + base);
        v4i* ldst = (v4i*)(smem + tid);
        __builtin_amdgcn_global_load_async_to_lds_b128(gsrc, ldst, 0 * TILE_BYTES, 0);
        __builtin_amdgcn_global_load_async_to_lds_b128(gsrc, ldst, 1 * TILE_BYTES, 0);
        __builtin_amdgcn_global_load_async_to_lds_b128(gsrc, ldst, 2 * TILE_BYTES, 0);
        __builtin_amdgcn_global_load_async_to_lds_b128(gsrc, ldst, 3 * TILE_BYTES, 0);
#if __has_builtin(__builtin_amdgcn_s_wait_asynccnt)
        __builtin_amdgcn_s_wait_asynccnt(0);
#else
        asm volatile("s_wait_asynccnt 0" ::: "memory");
#endif
#pragma unroll
        for (int i = 0; i < ITEMS; ++i) {
            v4f v = smem[i * THREADS + tid];
            __builtin_nontemporal_store(cumcos(v), out + base + i * THREADS);
        }
    } else {
        // Tail path (not taken for this problem size): per-row guards.
#pragma unroll
        for (int i = 0; i < ITEMS; ++i) {
            int idx = base + i * THREADS;
            if (idx < B) {
                __builtin_amdgcn_global_load_async_to_lds_b128(
                    (v4i*)(x + idx), (v4i*)(smem + i * THREADS + tid), 0, 0);
            }
        }
#if __has_builtin(__builtin_amdgcn_s_wait_asynccnt)
        __builtin_amdgcn_s_wait_asynccnt(0);
#else
        asm volatile("s_wait_asynccnt 0" ::: "memory");
#endif
#pragma unroll
        for (int i = 0; i < ITEMS; ++i) {
            int idx = base + i * THREADS;
            if (idx < B) {
                v4f v = smem[i * THREADS + tid];
                __builtin_nontemporal_store(cumcos(v), out + idx);
            }
        }
    }
#else
    // Fallback: direct b128 NT loads (still streaming-optimal).
#pragma unroll
    for (int i = 0; i < ITEMS; ++i) {
        int idx = base + i * THREADS;
        if (idx < B) {
            v4f v = __builtin_nontemporal_load(x + idx);
            __builtin_nontemporal_store(cumcos(v), out + idx);
        }
    }
#endif
}

extern "C" void kernel_launch(void* const* d_in, const int* in_sizes, int n_in,
                              void* d_out, int out_size, void* d_ws, size_t ws_size,
                              hipStream_t stream) {
    (void)n_in; (void)d_ws; (void)ws_size; (void)out_size;
    const v4f* x = (const v4f*)d_in[0];
    // d_in[1] (weight) is provably unused: RZ phases and CNOT permutations do
    // not alter computational-basis probabilities.
    v4f* out = (v4f*)d_out;
    const int B = in_sizes[0] / 4;                 // 524288 batch rows
    const int per_block = THREADS * ITEMS;         // 1024 rows per block
    const int blocks = (B + per_block - 1) / per_block;
    qexpz_kernel<<<blocks, THREADS, 0, stream>>>(x, out, B);
}